// RQBottleneck_43645457662420
// MI455X (gfx1250) — compile-verified
//
#include <hip/hip_runtime.h>
#include <hip/hip_bf16.h>

typedef float v2f __attribute__((ext_vector_type(2)));
typedef float v8f __attribute__((ext_vector_type(8)));

#define NVEC   16384        // rows (B*h*w)
#define EDIM   256          // embedding dim
#define NEMB   16384        // codebook entries
#define NDEP   4
#define BM     128
#define BN     128
#define KC     32

// ---------------- codebook row norms (all depths at once) ----------------
__global__ __launch_bounds__(256) void cnorm_kernel(const float* __restrict__ cbs,
                                                    float* __restrict__ cnorm) {
    const int wave = (blockIdx.x * 256 + threadIdx.x) >> 5;   // row id, 0..4*16384-1
    const int lane = threadIdx.x & 31;
    const float* r = cbs + (size_t)wave * EDIM;
    float s = 0.f;
    const int base = lane * 8;
    float4 d0 = *(const float4*)(r + base);
    float4 d1 = *(const float4*)(r + base + 4);
    s += d0.x*d0.x + d0.y*d0.y + d0.z*d0.z + d0.w*d0.w;
    s += d1.x*d1.x + d1.y*d1.y + d1.z*d1.z + d1.w*d1.w;
    #pragma unroll
    for (int m = 16; m; m >>= 1) s += __shfl_xor(s, m, 32);
    if (lane == 0) cnorm[wave] = s;
}

// ---------------- residual init: x (latent) -> res (code shape) ----------
__global__ __launch_bounds__(256) void init_res_kernel(const float* __restrict__ x,
                                                       float* __restrict__ res,
                                                       float* __restrict__ lossAcc) {
    const int i = blockIdx.x * 256 + threadIdx.x;    // over N*E = 4194304
    if (i == 0) *lossAcc = 0.f;
    const int n = i >> 8, e = i & 255;
    const int dd = e & 63, rw = (e >> 6) & 1, rh = (e >> 7) & 1;
    const int w2 = n & 15, h2 = (n >> 4) & 15, b = n >> 8;
    const int hh = h2 * 2 + rh, ww = w2 * 2 + rw;
    res[i] = x[((b * 32 + hh) * 32 + ww) * 64 + dd];
}

__global__ __launch_bounds__(256) void init_minp_kernel(unsigned long long* __restrict__ minp) {
    const int i = blockIdx.x * 256 + threadIdx.x;
    if (i < NVEC) minp[i] = ~0ull;
}

// ---------------- fused fp32 WMMA GEMM + per-row argmin ------------------
// 256 threads = 8 waves; wave (wm, wn) computes a 32x64 strip (8 16x16 C tiles).
// Double-buffered LDS: prefetch next K-chunk into registers during compute.
__global__ __launch_bounds__(256) void gemm_argmin_kernel(
    const float* __restrict__ res,      // [NVEC, EDIM]
    const float* __restrict__ cb,       // [NEMB, EDIM] (one depth)
    const float* __restrict__ cnorm,    // [NEMB]
    unsigned long long* __restrict__ minp) {

    __shared__ float As[2][BM][KC + 1];
    __shared__ float Bs[2][BN][KC + 1];

    const int tid  = threadIdx.x;
    const int lane = tid & 31;
    const int wave = tid >> 5;          // 0..7
    const int wm   = wave & 3;          // 4 waves along M (32 rows each)
    const int wn   = wave >> 2;         // 2 waves along N (64 cols each)
    const int half = lane >> 4;         // 0: lanes 0-15, 1: lanes 16-31
    const int lid  = lane & 15;
    const int rowBase = blockIdx.x * BM;
    const int colBase = blockIdx.y * BN;

    v8f acc[2][4] = {};

    // prologue: stage chunk 0 (128x32 A and B = 1024 float4 each, 4/thread)
    #pragma unroll
    for (int i = 0; i < 4; ++i) {
        const int v = tid + i * 256;
        const int r = v >> 3;                 // 8 float4 per row of 32
        const int c = (v & 7) << 2;
        const float4 da = *(const float4*)(res + (size_t)(rowBase + r) * EDIM + c);
        As[0][r][c] = da.x; As[0][r][c+1] = da.y; As[0][r][c+2] = da.z; As[0][r][c+3] = da.w;
        const float4 db = *(const float4*)(cb + (size_t)(colBase + r) * EDIM + c);
        Bs[0][r][c] = db.x; Bs[0][r][c+1] = db.y; Bs[0][r][c+2] = db.z; Bs[0][r][c+3] = db.w;
    }
    __syncthreads();

    const int nchunks = EDIM / KC;      // 8
    for (int kc = 0; kc < nchunks; ++kc) {
        const int cur = kc & 1, nxt = cur ^ 1;
        const bool more = (kc + 1 < nchunks);

        float4 pa[4], pb[4];
        if (more) {
            const int k0 = (kc + 1) * KC;
            #pragma unroll
            for (int i = 0; i < 4; ++i) {
                const int v = tid + i * 256;
                const int r = v >> 3;
                const int c = (v & 7) << 2;
                pa[i] = *(const float4*)(res + (size_t)(rowBase + r) * EDIM + k0 + c);
                pb[i] = *(const float4*)(cb  + (size_t)(colBase + r) * EDIM + k0 + c);
            }
        }

        #pragma unroll
        for (int ks = 0; ks < KC; ks += 4) {
            // A 16x4 fp32 frags: lanes0-15 hold K=ks,ks+1; lanes16-31 K=ks+2,ks+3
            v2f a0, a1;
            a0.x = As[cur][wm * 32 +      lid][ks + half * 2 + 0];
            a0.y = As[cur][wm * 32 +      lid][ks + half * 2 + 1];
            a1.x = As[cur][wm * 32 + 16 + lid][ks + half * 2 + 0];
            a1.y = As[cur][wm * 32 + 16 + lid][ks + half * 2 + 1];
            #pragma unroll
            for (int t = 0; t < 4; ++t) {
                v2f b;   // B 4x16: N=lid across lanes, K split like A
                b.x = Bs[cur][wn * 64 + t * 16 + lid][ks + half * 2 + 0];
                b.y = Bs[cur][wn * 64 + t * 16 + lid][ks + half * 2 + 1];
                acc[0][t] = __builtin_amdgcn_wmma_f32_16x16x4_f32(
                    false, a0, false, b, (short)0, acc[0][t], false, false);
                acc[1][t] = __builtin_amdgcn_wmma_f32_16x16x4_f32(
                    false, a1, false, b, (short)0, acc[1][t], false, false);
            }
        }

        if (more) {
            #pragma unroll
            for (int i = 0; i < 4; ++i) {
                const int v = tid + i * 256;
                const int r = v >> 3;
                const int c = (v & 7) << 2;
                As[nxt][r][c] = pa[i].x; As[nxt][r][c+1] = pa[i].y;
                As[nxt][r][c+2] = pa[i].z; As[nxt][r][c+3] = pa[i].w;
                Bs[nxt][r][c] = pb[i].x; Bs[nxt][r][c+1] = pb[i].y;
                Bs[nxt][r][c+2] = pb[i].z; Bs[nxt][r][c+3] = pb[i].w;
            }
            __syncthreads();
        }
    }

    // epilogue: d = ||c_j||^2 - 2 * <res_i, c_j>; argmin via packed u64 atomic min
    const int myCol = colBase + wn * 64;
    #pragma unroll
    for (int m = 0; m < 2; ++m) {
        #pragma unroll
        for (int v = 0; v < 8; ++v) {
            unsigned long long best = ~0ull;
            #pragma unroll
            for (int t = 0; t < 4; ++t) {
                const int col = myCol + t * 16 + lid;
                const float dist = cnorm[col] - 2.0f * acc[m][t][v];
                unsigned u = __float_as_uint(dist);
                u = (u & 0x80000000u) ? ~u : (u | 0x80000000u);   // monotone key
                const unsigned long long p = ((unsigned long long)u << 32) | (unsigned)col;
                best = (p < best) ? p : best;
            }
            // min across the 16 lanes of this half (rows ... v / v+8)
            #pragma unroll
            for (int s = 1; s <= 8; s <<= 1) {
                const unsigned long long o = __shfl_xor(best, s, 32);
                best = (o < best) ? o : best;
            }
            if (lid == 0) {
                const int row = rowBase + wm * 32 + m * 16 + v + half * 8;
                atomicMin(minp + row, best);
            }
        }
    }
}

// ---------------- residual update + loss + codes -------------------------
__global__ __launch_bounds__(256) void update_kernel(
    const float* __restrict__ cb, const unsigned long long* __restrict__ minp,
    float* __restrict__ res, float* __restrict__ codesOut,
    float* __restrict__ lossAcc, int depth) {

    __shared__ float sred[8];
    const int n = blockIdx.x;
    const int e = threadIdx.x;
    const unsigned long long p = minp[n];
    const int code = (int)(unsigned)(p & 0xffffffffull);
    const float q = cb[(size_t)code * EDIM + e];
    const float r = res[(size_t)n * EDIM + e] - q;
    res[(size_t)n * EDIM + e] = r;

    float sq = r * r;
    #pragma unroll
    for (int m = 16; m; m >>= 1) sq += __shfl_xor(sq, m, 32);
    if ((threadIdx.x & 31) == 0) sred[threadIdx.x >> 5] = sq;
    __syncthreads();
    if (threadIdx.x == 0) {
        float s = 0.f;
        #pragma unroll
        for (int i = 0; i < 8; ++i) s += sred[i];
        atomicAdd(lossAcc, s);
        codesOut[n * NDEP + depth] = (float)code;
    }
}

// ---------------- out = x - res (latent layout), loss scalar -------------
__global__ __launch_bounds__(256) void finalize_kernel(
    const float* __restrict__ x, const float* __restrict__ res,
    float* __restrict__ out, float* __restrict__ lossOut,
    const float* __restrict__ lossAcc) {
    const int i = blockIdx.x * 256 + threadIdx.x;   // latent linear index
    const int dd = i & 63, ww = (i >> 6) & 31, hh = (i >> 11) & 31, b = i >> 16;
    const int h2 = hh >> 1, rh = hh & 1, w2 = ww >> 1, rw = ww & 1;
    const int n = (b * 16 + h2) * 16 + w2;
    const int e = ((rh * 2 + rw) << 6) | dd;
    out[i] = x[i] - res[(size_t)n * EDIM + e];
    if (i == 0) *lossOut = *lossAcc / (float)(NDEP) / (float)NVEC / (float)EDIM;
}

extern "C" void kernel_launch(void* const* d_in, const int* in_sizes, int n_in,
                              void* d_out, int out_size, void* d_ws, size_t ws_size,
                              hipStream_t stream) {
    const float* x   = (const float*)d_in[0];               // [64,32,32,64]
    const float* cbs = (const float*)d_in[1];               // [4,16384,256]
    float* out      = (float*)d_out;                        // 4194304 floats
    float* lossOut  = out + 4194304;                        // 1 float
    float* codesOut = out + 4194305;                        // 65536 entries

    char* ws = (char*)d_ws;
    float* res = (float*)ws;                                              // 16 MB
    float* cnorm = (float*)(ws + (size_t)16 * 1024 * 1024);               // 1 MB region
    unsigned long long* minp =
        (unsigned long long*)(ws + (size_t)16 * 1024 * 1024 + 1048576);   // 128 KB
    float* lossAcc = (float*)(ws + (size_t)16 * 1024 * 1024 + 1048576 + 131072);

    // codebook norms for all depths: 4*16384 rows, 1 wave each, 8 waves/block
    cnorm_kernel<<<(NDEP * NEMB) / 8, 256, 0, stream>>>(cbs, cnorm);
    // residual init
    init_res_kernel<<<(NVEC * EDIM) / 256, 256, 0, stream>>>(x, res, lossAcc);

    const dim3 ggrid(NVEC / BM, NEMB / BN);
    for (int d = 0; d < NDEP; ++d) {
        const float* cb = cbs + (size_t)d * NEMB * EDIM;
        init_minp_kernel<<<NVEC / 256, 256, 0, stream>>>(minp);
        gemm_argmin_kernel<<<ggrid, 256, 0, stream>>>(res, cb, cnorm + (size_t)d * NEMB, minp);
        update_kernel<<<NVEC, 256, 0, stream>>>(cb, minp, res, codesOut, lossAcc, d);
    }
    finalize_kernel<<<(64 * 32 * 32 * 64) / 256, 256, 0, stream>>>(x, res, out, lossOut, lossAcc);
}